// _HSA_decode_15547781612186
// MI455X (gfx1250) — compile-verified
//
#include <hip/hip_runtime.h>
#include <stdint.h>

typedef float    v2f   __attribute__((ext_vector_type(2)));
typedef float    v8f   __attribute__((ext_vector_type(8)));
typedef uint32_t u32x4 __attribute__((ext_vector_type(4)));
typedef int      i32x4 __attribute__((ext_vector_type(4)));
typedef int      i32x8 __attribute__((ext_vector_type(8)));

#define NB    16
#define NHQ   32
#define NH    8
#define ND    128
#define NG    4
#define MAXS  8192
#define BLKS  64
#define NS    32
#define SPLIT 4
#define BLK_PER_WG (NS / SPLIT)

#define KROW 132   // 128 + 4 pad dwords  -> conflict-free A-fragment (K) reads
#define VROW 136   // 128 + 8 pad dwords  -> conflict-free B-fragment (V) reads
#define QROW 132   // 128 + 4 pad dwords  -> conflict-free q B-fragment reads

__device__ __forceinline__ uint32_t lds_offset(const void* p) {
  // LDS flat aperture: low 32 bits of a shared-memory address are the LDS byte offset
  return (uint32_t)(uintptr_t)p;
}

// Issue one TDM 2-D tile load: 64 rows x 128 f32, row stride = H*D = 1024 f32,
// written to LDS with `pad` dwords appended after every 128-dword row.
__device__ __forceinline__ void tdm_load_tile(uint32_t lds_byte, const float* gptr,
                                              uint32_t pad_amount_code) {
  uint64_t ga = (uint64_t)(uintptr_t)gptr;
  u32x4 g0;
  g0[0] = 1u;                                                  // count = 1 valid descriptor
  g0[1] = lds_byte;                                            // LDS destination (bytes)
  g0[2] = (uint32_t)ga;                                        // global addr lo
  g0[3] = ((uint32_t)(ga >> 32) & 0x01FFFFFFu) | (2u << 30);   // global addr hi | type=2
  i32x8 g1;
  g1[0] = (int)((2u << 16)                 // data_size = 4 bytes
              | (1u << 20)                 // pad_enable
              | (6u << 22)                 // pad_interval: every 128 dwords (one row)
              | (pad_amount_code << 25));  // pad_amount code (3 -> 4 dw, 7 -> 8 dw)
  g1[1] = (int)(128u << 16);               // tensor_dim0 = 128   (bits 79:48)
  g1[2] = (int)(64u  << 16);               // tensor_dim1 = 64    (bits 111:80)
  g1[3] = (int)(128u << 16);               // tile_dim0  = 128    (bits 127:112)
  g1[4] = (int)64;                         // tile_dim1  = 64     (bits 143:128)
  g1[5] = (int)(NH * ND);                  // tensor_dim0_stride = 1024 (bits 207:160)
  g1[6] = 0;
  g1[7] = 0;
  i32x4 z4 = {0, 0, 0, 0};
#if defined(__clang_major__) && __clang_major__ >= 23
  i32x8 z8 = {0, 0, 0, 0, 0, 0, 0, 0};
  __builtin_amdgcn_tensor_load_to_lds(g0, g1, z4, z4, z8, 0);
#else
  __builtin_amdgcn_tensor_load_to_lds(g0, g1, z4, z4, 0);
#endif
}

__global__ void zero_out_kernel(float* o, int n) {
  int i = blockIdx.x * blockDim.x + threadIdx.x;
  if (i < n) o[i] = 0.0f;
}

__global__ __launch_bounds__(128) void sparse_decode_kernel(
    const float* __restrict__ qg, const float* __restrict__ kg,
    const float* __restrict__ vg, const float* __restrict__ wg,
    const int* __restrict__ bidx, float* __restrict__ outg)
{
  extern __shared__ float smem[];
  float* kbuf = smem;                          // 2 * 64 * KROW
  float* vbuf = kbuf + 2 * BLKS * KROW;        // 2 * 64 * VROW
  float* qs   = vbuf + 2 * BLKS * VROW;        // 4 * QROW (scaled q)
  float* exm  = qs + NG * QROW;                // 4 waves * 16
  float* exs  = exm + 64;                      // 4 waves * 16

  const int tid   = threadIdx.x;
  const int lane  = tid & 31;
  const int wv    = tid >> 5;        // wave id 0..3 -> owns keys [16*wv, 16*wv+16)
  const int lr    = lane & 15;
  const int hs    = lane >> 4;       // wmma half-select (K-operand pairing)
  const int wgid  = blockIdx.x;
  const int split = wgid & (SPLIT - 1);
  const int bh    = wgid / SPLIT;
  const int b     = bh / NH;
  const int h     = bh % NH;
  const int s0    = split * BLK_PER_WG;

  const float scale = 0.088388347648318447f;   // 1/sqrt(128)

  // stage q (pre-scaled) into padded LDS rows
  for (int i = tid; i < NG * ND; i += 128) {
    int g = i >> 7, d = i & (ND - 1);
    qs[g * QROW + d] = qg[((size_t)b * NHQ + h * NG + g) * ND + d] * scale;
  }

  v8f acc[8];
#pragma unroll
  for (int n = 0; n < 8; ++n)
#pragma unroll
    for (int r = 0; r < 8; ++r) acc[n][r] = 0.0f;

  // prefetch block 0 (wave 0 drives the DMA engine)
  if (wv == 0) {
    int idx = __builtin_amdgcn_readfirstlane(bidx[bh * NS + s0]);
    if (idx < 0) idx = 0;
    const float* kp = kg + (((size_t)b * MAXS + (size_t)idx * BLKS) * NH + h) * ND;
    const float* vp = vg + (((size_t)b * MAXS + (size_t)idx * BLKS) * NH + h) * ND;
    tdm_load_tile(lds_offset(kbuf), kp, 3u);
    tdm_load_tile(lds_offset(vbuf), vp, 7u);
  }

  for (int si = 0; si < BLK_PER_WG; ++si) {
    const int s   = s0 + si;
    const int cur = si & 1;
    const float* kb = kbuf + cur * BLKS * KROW;
    const float* vb = vbuf + cur * BLKS * VROW;

    if (wv == 0) {
      if (si + 1 < BLK_PER_WG) {
        const int nxt = cur ^ 1;
        int idx = __builtin_amdgcn_readfirstlane(bidx[bh * NS + s + 1]);
        if (idx < 0) idx = 0;
        const float* kp = kg + (((size_t)b * MAXS + (size_t)idx * BLKS) * NH + h) * ND;
        const float* vp = vg + (((size_t)b * MAXS + (size_t)idx * BLKS) * NH + h) * ND;
        tdm_load_tile(lds_offset(kbuf + nxt * BLKS * KROW), kp, 3u);
        tdm_load_tile(lds_offset(vbuf + nxt * BLKS * VROW), vp, 7u);
        __builtin_amdgcn_s_wait_tensorcnt(2);  // current buffer done, prefetch in flight
      } else {
        __builtin_amdgcn_s_wait_tensorcnt(0);
      }
    }
    __syncthreads();   // buf[cur] ready for all waves (also orders qs staging, iter 0)

    // ---- scores_T tile: D[key][group] = sum_d K[key][d] * q[group][d]*scale ----
    v8f sc;
#pragma unroll
    for (int r = 0; r < 8; ++r) sc[r] = 0.0f;

    const float* krow = kb + (16 * wv + lr) * KROW + 2 * hs;
    const float* qrow = qs + (lr & 3) * QROW + 2 * hs;   // groups>3 are ignored dup cols
#pragma unroll
    for (int u = 0; u < 32; ++u) {
      v2f a  = *(const v2f*)(krow + 4 * u);
      v2f bq = *(const v2f*)(qrow + 4 * u);
      sc = __builtin_amdgcn_wmma_f32_16x16x4_f32(false, a, false, bq,
                                                 (short)0, sc, false, false);
    }

    // ---- per-block softmax over the 64 keys (combine 4 waves via LDS) ----
    float mloc = sc[0];
#pragma unroll
    for (int r = 1; r < 8; ++r) mloc = fmaxf(mloc, sc[r]);
    mloc = fmaxf(mloc, __shfl_xor(mloc, 16, 32));
    if (lane < 4) exm[wv * 16 + lane] = mloc;
    __syncthreads();
    const int gcl = lr & 3;
    float M = fmaxf(fmaxf(exm[gcl], exm[16 + gcl]),
                    fmaxf(exm[32 + gcl], exm[48 + gcl]));
    float ssum = 0.0f;
#pragma unroll
    for (int r = 0; r < 8; ++r) { sc[r] = __expf(sc[r] - M); ssum += sc[r]; }
    ssum += __shfl_xor(ssum, 16, 32);
    if (lane < 4) exs[wv * 16 + lane] = ssum;
    __syncthreads();
    float SUM = exs[gcl] + exs[16 + gcl] + exs[32 + gcl] + exs[48 + gcl];

    int   rawidx = bidx[bh * NS + s];
    float wt   = wg[((size_t)b * NHQ + h * NG + gcl) * NS + s];
    float coef = (rawidx >= 0) ? (wt / SUM) : 0.0f;
#pragma unroll
    for (int r = 0; r < 8; ++r) sc[r] *= coef;

    // ---- out += P(4x16) @ V(16x128): P's D-layout -> A-layout via half-swaps ----
    const float* vrow = vb + (16 * wv) * VROW + lr;
#pragma unroll
    for (int u = 0; u < 4; ++u) {
      float a0, a1;
      if (u < 2) {
        float o0 = __shfl_xor(sc[4 * u + 2], 16, 32);
        float o1 = __shfl_xor(sc[4 * u + 3], 16, 32);
        a0 = (lane < 16) ? sc[4 * u]     : o0;
        a1 = (lane < 16) ? sc[4 * u + 1] : o1;
      } else {
        float o0 = __shfl_xor(sc[4 * u - 8], 16, 32);
        float o1 = __shfl_xor(sc[4 * u - 7], 16, 32);
        a0 = (lane < 16) ? o0 : sc[4 * u - 6];
        a1 = (lane < 16) ? o1 : sc[4 * u - 5];
      }
      v2f af; af.x = a0; af.y = a1;
      const int row0 = 4 * u + 2 * hs;
#pragma unroll
      for (int n = 0; n < 8; ++n) {
        v2f bf;
        bf.x = vrow[row0 * VROW + n * 16];
        bf.y = vrow[(row0 + 1) * VROW + n * 16];
        acc[n] = __builtin_amdgcn_wmma_f32_16x16x4_f32(false, af, false, bf,
                                                       (short)0, acc[n], false, false);
      }
    }
    __syncthreads();   // release buf[cur] before next iteration's prefetch overwrites it
  }

  // valid output lives in lanes 0-15 (N=d), VGPRs 0-3 (M=group)
  if (lane < 16) {
#pragma unroll
    for (int n = 0; n < 8; ++n)
#pragma unroll
      for (int r = 0; r < 4; ++r)
        atomicAdd(&outg[((size_t)b * NHQ + h * NG + r) * ND + n * 16 + lane],
                  acc[n][r]);
  }
}

extern "C" void kernel_launch(void* const* d_in, const int* in_sizes, int n_in,
                              void* d_out, int out_size, void* d_ws, size_t ws_size,
                              hipStream_t stream) {
  const float* q  = (const float*)d_in[0];
  const float* k  = (const float*)d_in[1];
  const float* v  = (const float*)d_in[2];
  const float* w  = (const float*)d_in[3];
  const int*   bi = (const int*)d_in[4];
  float* out = (float*)d_out;

  zero_out_kernel<<<(out_size + 255) / 256, 256, 0, stream>>>(out, out_size);

  size_t shmem = (size_t)(2 * BLKS * KROW + 2 * BLKS * VROW + NG * QROW + 128) * sizeof(float);
  sparse_decode_kernel<<<NB * NH * SPLIT, 128, shmem, stream>>>(q, k, v, w, bi, out);
  (void)in_sizes; (void)n_in; (void)d_ws; (void)ws_size;
}